// Net_time_no_conv_6648609374271
// MI455X (gfx1250) — compile-verified
//
#include <hip/hip_runtime.h>

// CDNA5 / gfx1250. Exploits the structural collapse of the SNN: the 2-input
// heaviside front-end means every batch row maps to one of 4 spike patterns,
// so the whole network is a 4-entry LUT. Kernel 1 builds the LUT with exact
// fp32 arithmetic using V_WMMA_F32_16X16X4_F32 (4 patterns padded to M=16).
// Kernel 2 is the HBM-bound scatter (~15 MB traffic -> ~0.65 us @ 23.3 TB/s).

typedef __attribute__((ext_vector_type(2))) float v2f;
typedef __attribute__((ext_vector_type(8))) float v8f;

#define LDS_STRIDE 448  // max feature width (layer-3 output), fp32

// One dense layer: act_out[16][Nout] = heaviside(act_in[16][Kin] * W^T + b - th)
// W is row-major [Nout][Kin] (torch Linear layout). Waves split N-tiles.
__device__ __forceinline__ void wmma_layer(
    const float* __restrict__ ldsIn, float* __restrict__ ldsOut,
    const float* __restrict__ W, const float* __restrict__ bias,
    int Kin, int Nout, float th, int lane, int wave, int nwaves)
{
    const int n  = lane & 15;        // B/D column within tile
    const int m  = lane & 15;        // A row (pattern index)
    const int hi = lane >> 4;
    const int kb = hi * 2;           // K sub-offset for this half-wave
    const int ntiles = (Nout + 15) >> 4;
    const int ktiles = (Kin  +  3) >> 2;

    for (int nt = wave; nt < ntiles; nt += nwaves) {
        const int col = nt * 16 + n;
        const bool cok = (col < Nout);
        v8f acc = {};
        for (int kt = 0; kt < ktiles; ++kt) {
            const int k0 = kt * 4 + kb;
            v2f a, b;
            a.x = ldsIn[m * LDS_STRIDE + k0];
            a.y = ldsIn[m * LDS_STRIDE + k0 + 1];
            b.x = (cok && (k0     < Kin)) ? W[col * Kin + k0]     : 0.0f;
            b.y = (cok && (k0 + 1 < Kin)) ? W[col * Kin + k0 + 1] : 0.0f;
            acc = __builtin_amdgcn_wmma_f32_16x16x4_f32(
                false, a, false, b, (short)0, acc, false, false);
        }
        const float bv = cok ? bias[col] : 0.0f;
#pragma unroll
        for (int r = 0; r < 8; ++r) {
            const int row = r + 8 * hi;               // D row = pattern index
            const float cur = acc[r] + bv;
            const float spk = ((cur - th) > 0.0f) ? 1.0f : 0.0f;
            ldsOut[row * LDS_STRIDE + col] = cok ? spk : 0.0f;  // zero-pad tail
        }
    }
}

extern "C" __global__ void __launch_bounds__(256)
snn_table_kernel(const float* __restrict__ W0, const float* __restrict__ b0,
                 const float* __restrict__ W1, const float* __restrict__ b1,
                 const float* __restrict__ W2, const float* __restrict__ b2,
                 const float* __restrict__ W3, const float* __restrict__ b3,
                 const float* __restrict__ th1p, const float* __restrict__ th2p,
                 const float* __restrict__ th3p, const float* __restrict__ th4p,
                 float* __restrict__ table /* [0..23]=spk[4][6], [24..47]=mem[4][6] */)
{
    __shared__ float actA[16 * LDS_STRIDE];
    __shared__ float actB[16 * LDS_STRIDE];
    const int tid  = threadIdx.x;
    const int lane = tid & 31;
    const int wave = tid >> 5;

    // Input spikes: 16 pattern rows x (K padded to 4). S[p][k] = (p>>k)&1, k<2.
    if (tid < 64) {
        const int mm = tid >> 2, k = tid & 3;
        actA[mm * LDS_STRIDE + k] = (k < 2) ? (float)((mm >> k) & 1) : 0.0f;
    }
    __syncthreads();

    wmma_layer(actA, actB, W0, b0,   2,  40, *th1p, lane, wave, 8);
    __syncthreads();
    wmma_layer(actB, actA, W1, b1,  40, 216, *th2p, lane, wave, 8);
    __syncthreads();
    wmma_layer(actA, actB, W2, b2, 216, 448, *th3p, lane, wave, 8);
    __syncthreads();

    // Final layer: Nout = 6 (1 tile), wave 0 only; also capture mem3 = cur3.
    if (wave == 0) {
        const float th4 = *th4p;
        const int n  = lane & 15;
        const int m  = lane & 15;
        const int hi = lane >> 4;
        const int kb = hi * 2;
        v8f acc = {};
        for (int kt = 0; kt < 112; ++kt) {   // Kin = 448
            const int k0 = kt * 4 + kb;
            v2f a, b;
            a.x = actB[m * LDS_STRIDE + k0];
            a.y = actB[m * LDS_STRIDE + k0 + 1];
            b.x = (n < 6) ? W3[n * 448 + k0]     : 0.0f;
            b.y = (n < 6) ? W3[n * 448 + k0 + 1] : 0.0f;
            acc = __builtin_amdgcn_wmma_f32_16x16x4_f32(
                false, a, false, b, (short)0, acc, false, false);
        }
        if (n < 6) {
            const float bv = b3[n];
#pragma unroll
            for (int r = 0; r < 8; ++r) {
                const int row = r + 8 * hi;
                if (row < 4) {
                    const float cur = acc[r] + bv;
                    table[row * 6 + n]      = ((cur - th4) > 0.0f) ? 1.0f : 0.0f;
                    table[24 + row * 6 + n] = cur;
                }
            }
        }
    }
}

extern "C" __global__ void __launch_bounds__(256)
snn_scatter_kernel(const float* __restrict__ data_in,
                   const float* __restrict__ th0,
                   const float* __restrict__ table,
                   float* __restrict__ out, int B)
{
    __shared__ float t[48];
    if (threadIdx.x < 48) t[threadIdx.x] = table[threadIdx.x];
    __syncthreads();

    const int i = blockIdx.x * blockDim.x + threadIdx.x;
    if (i >= B) return;

    const float t0 = th0[0], t1 = th0[1];
    const float2 d = reinterpret_cast<const float2*>(data_in)[i];
    const int p = (int)(d.x > t0) | ((int)(d.y > t1) << 1);

    const float* ts = &t[p * 6];
    const float* tm = &t[24 + p * 6];

    // 24-byte row stride: 8-byte aligned -> three b64 stores per half.
    float2* so = reinterpret_cast<float2*>(out) + (size_t)i * 3;
    float2* mo = reinterpret_cast<float2*>(out + (size_t)B * 6) + (size_t)i * 3;
    so[0] = make_float2(ts[0], ts[1]);
    so[1] = make_float2(ts[2], ts[3]);
    so[2] = make_float2(ts[4], ts[5]);
    mo[0] = make_float2(tm[0], tm[1]);
    mo[1] = make_float2(tm[2], tm[3]);
    mo[2] = make_float2(tm[4], tm[5]);
}

extern "C" void kernel_launch(void* const* d_in, const int* in_sizes, int n_in,
                              void* d_out, int out_size, void* d_ws, size_t ws_size,
                              hipStream_t stream) {
    const float* data_in = (const float*)d_in[0];
    const float* W0 = (const float*)d_in[1];
    const float* b0 = (const float*)d_in[2];
    const float* W1 = (const float*)d_in[3];
    const float* b1 = (const float*)d_in[4];
    const float* W2 = (const float*)d_in[5];
    const float* b2 = (const float*)d_in[6];
    const float* W3 = (const float*)d_in[7];
    const float* b3 = (const float*)d_in[8];
    const float* th0 = (const float*)d_in[9];
    const float* th1 = (const float*)d_in[10];
    const float* th2 = (const float*)d_in[11];
    const float* th3 = (const float*)d_in[12];
    const float* th4 = (const float*)d_in[13];
    // d_in[14..17] = beta1..4, d_in[18..21] = V0..V3: algebraically unused
    // (prior mem/spk are zero).

    float* table = (float*)d_ws;  // 48 floats
    const int B = in_sizes[0] / 2;

    snn_table_kernel<<<1, 256, 0, stream>>>(W0, b0, W1, b1, W2, b2, W3, b3,
                                            th1, th2, th3, th4, table);
    snn_scatter_kernel<<<(B + 255) / 256, 256, 0, stream>>>(
        data_in, th0, table, (float*)d_out, B);
}